// HRTExtractor_81320910782627
// MI455X (gfx1250) — compile-verified
//
#include <hip/hip_runtime.h>
#include <math.h>

// Problem constants (from reference setup_inputs)
constexpr int N_DOC = 4;
constexpr int L     = 1024;
constexpr int D     = 768;
constexpr int H     = 12;
constexpr int E     = 32;
constexpr int M     = 4;
constexpr int R     = 256;
constexpr float NEG_MAX = -3.402823466e+38f;

typedef __attribute__((ext_vector_type(2))) float v2f;
typedef __attribute__((ext_vector_type(4))) float v4f;
typedef __attribute__((ext_vector_type(8))) float v8f;

// ---------------------------------------------------------------------------
// Kernel 1: e_emb[i,e,c] = logsumexp_m( mask>0 ? seq[i, pos[i,e,m]+1, c] : -MAX )
// grid = N_DOC*E blocks, 256 threads
// ---------------------------------------------------------------------------
__global__ void k_e_emb(const float* __restrict__ seq,
                        const int*   __restrict__ pos,
                        const float* __restrict__ mask,
                        float* __restrict__ e_emb) {
    int b = blockIdx.x;
    int i = b / E;
    const float* seq_i  = seq  + (size_t)i * L * D;
    const int*   pos_e  = pos  + (size_t)b * M;
    const float* mask_e = mask + (size_t)b * M;

    int   p[M];
    float mk[M];
#pragma unroll
    for (int m = 0; m < M; ++m) { p[m] = pos_e[m] + 1; mk[m] = mask_e[m]; }

    for (int c = threadIdx.x; c < D; c += blockDim.x) {
        float v[M];
        float mx = NEG_MAX;
#pragma unroll
        for (int m = 0; m < M; ++m) {
            float val = seq_i[(size_t)p[m] * D + c];
            v[m] = (mk[m] > 0.0f) ? val : NEG_MAX;
            mx = fmaxf(mx, v[m]);
        }
        float s = 0.0f;
#pragma unroll
        for (int m = 0; m < M; ++m)
            if (mk[m] > 0.0f) s += expf(v[m] - mx);
        e_emb[(size_t)b * D + c] = mx + logf(s);
    }
}

// ---------------------------------------------------------------------------
// Kernel 2: e_att[i,e,hh,l] = sum_m mask * att[i,hh,pos+1,l] / max(cnt,1)
// grid = N_DOC*E*H blocks, 256 threads x float4 (256*4 = 1024 = L)
// ---------------------------------------------------------------------------
__global__ void k_e_att(const float* __restrict__ att,
                        const int*   __restrict__ pos,
                        const float* __restrict__ mask,
                        float* __restrict__ e_att) {
    int b  = blockIdx.x;
    int hh = b % H;
    int e  = (b / H) % E;
    int i  = b / (H * E);

    const int*   pos_e  = pos  + ((size_t)i * E + e) * M;
    const float* mask_e = mask + ((size_t)i * E + e) * M;

    float cnt = 0.0f;
#pragma unroll
    for (int m = 0; m < M; ++m) cnt += mask_e[m];
    float inv = 1.0f / fmaxf(cnt, 1.0f);

    int l0 = threadIdx.x * 4;
    v4f acc = {0.0f, 0.0f, 0.0f, 0.0f};
#pragma unroll
    for (int m = 0; m < M; ++m) {
        float mk = mask_e[m];
        if (mk > 0.0f) {
            int p = pos_e[m] + 1;
            const v4f* row = (const v4f*)(att + (((size_t)i * H + hh) * L + p) * L + l0);
            acc += mk * row[0];
        }
    }
    acc *= inv;
    v4f* dst = (v4f*)(e_att + (((size_t)i * E + e) * H + hh) * L + l0);
    dst[0] = acc;
}

// ---------------------------------------------------------------------------
// Kernel 3: ht_att[i,r,l] = normalize_l( mean_h( e_att[i,eh,:,l]*e_att[i,et,:,l] ) )
// grid = N_DOC*R blocks, 256 threads x float4
// ---------------------------------------------------------------------------
__global__ void k_ht_att(const float* __restrict__ e_att,
                         const int*   __restrict__ hts,
                         float* __restrict__ ht_att) {
    int b = blockIdx.x;             // i*R + r
    int i = b / R;
    int eh = hts[(size_t)b * 2 + 0];
    int et = hts[(size_t)b * 2 + 1];

    const float* ha = e_att + ((size_t)i * E + eh) * H * L;
    const float* ta = e_att + ((size_t)i * E + et) * H * L;

    int l0 = threadIdx.x * 4;
    v4f acc = {0.0f, 0.0f, 0.0f, 0.0f};
#pragma unroll
    for (int hh = 0; hh < H; ++hh) {
        v4f a = *(const v4f*)(ha + (size_t)hh * L + l0);
        v4f t = *(const v4f*)(ta + (size_t)hh * L + l0);
        acc += a * t;
    }
    acc *= (1.0f / (float)H);

    __shared__ float red[256];
    float ps = acc.x + acc.y + acc.z + acc.w;
    red[threadIdx.x] = ps;
    __syncthreads();
    for (int s = 128; s > 0; s >>= 1) {
        if (threadIdx.x < s) red[threadIdx.x] += red[threadIdx.x + s];
        __syncthreads();
    }
    float invs = 1.0f / (red[0] + 1e-5f);

    v4f* dst = (v4f*)(ht_att + (size_t)b * L + l0);
    dst[0] = acc * invs;
}

// ---------------------------------------------------------------------------
// Kernel 4: hs/ts gathers into out sections 0 and 1
// grid = N_DOC*R blocks, 256 threads
// ---------------------------------------------------------------------------
__global__ void k_hs_ts(const float* __restrict__ e_emb,
                        const int*   __restrict__ hts,
                        float* __restrict__ out) {
    int b = blockIdx.x;             // i*R + r
    int i = b / R;
    int eh = hts[(size_t)b * 2 + 0];
    int et = hts[(size_t)b * 2 + 1];

    const float* sh = e_emb + ((size_t)i * E + eh) * D;
    const float* st = e_emb + ((size_t)i * E + et) * D;
    float* oh = out + (size_t)b * D;                               // section 0 (hs)
    float* ot = out + (size_t)N_DOC * R * D + (size_t)b * D;       // section 1 (ts)

    for (int c = threadIdx.x; c < D; c += blockDim.x) {
        oh[c] = sh[c];
        ot[c] = st[c];
    }
}

// ---------------------------------------------------------------------------
// Kernel 5: rs = ht_att (R x L) @ seq (L x D) per doc, via V_WMMA_F32_16X16X4_F32
// Register-tiled: each wave32 computes a 32x16 output tile (2 M-tiles) so the
// strided B fragment (seq) is reused across two WMMAs -> 2 loads/WMMA.
// grid = N_DOC * (R/32) * (D/16) = 4*8*48 = 1536 waves.
// A 16x4 layout: lanes 0-15 -> M=lane, VGPR0/1 = K {k0,k0+1}; lanes 16-31 -> K {k0+2,k0+3}
// B 4x16 layout: lanes 0-15 -> N=lane, VGPR0/1 = K {k0,k0+1}; lanes 16-31 -> K {k0+2,k0+3}
// C/D 16x16: VGPR v -> M = v + 8*(lane>=16), N = lane&15
// ---------------------------------------------------------------------------
__global__ void k_rs_wmma(const float* __restrict__ seq,
                          const float* __restrict__ ht_att,
                          float* __restrict__ out) {
    constexpr int NT = D / 16;   // 48
    constexpr int MT = R / 32;   // 8 (32 rows per wave)
    int b  = blockIdx.x;
    int nt = b % NT;
    int mt = (b / NT) % MT;
    int i  = b / (NT * MT);

    int m0 = mt * 32;
    int n0 = nt * 16;

    const float* A  = ht_att + (size_t)i * R * L;   // (R, L) row-major
    const float* Bm = seq    + (size_t)i * L * D;   // (L, D) row-major

    int lane = threadIdx.x & 31;
    int half = lane >> 4;           // 0: K pair {k0,k0+1}, 1: K pair {k0+2,k0+3}
    int l15  = lane & 15;
    int bcol = n0 + l15;

    v8f c0 = {0.f, 0.f, 0.f, 0.f, 0.f, 0.f, 0.f, 0.f};
    v8f c1 = {0.f, 0.f, 0.f, 0.f, 0.f, 0.f, 0.f, 0.f};

    const float* arow0 = A + (size_t)(m0 + l15) * L;        // M-tile 0
    const float* arow1 = A + (size_t)(m0 + 16 + l15) * L;   // M-tile 1

    for (int k0 = 0; k0 < L; k0 += 4) {
        int ka = k0 + half * 2;
        v2f a0 = *(const v2f*)(arow0 + ka);         // contiguous K pair
        v2f a1 = *(const v2f*)(arow1 + ka);
        v2f bb;
        bb.x = Bm[(size_t)ka * D + bcol];
        bb.y = Bm[(size_t)(ka + 1) * D + bcol];
        c0 = __builtin_amdgcn_wmma_f32_16x16x4_f32(
                false, a0, false, bb, (short)0, c0, false, false);
        c1 = __builtin_amdgcn_wmma_f32_16x16x4_f32(
                false, a1, false, bb, (short)0, c1, false, false);
    }

    float* O = out + (size_t)2 * N_DOC * R * D + (size_t)i * R * D;   // section 2 (rs)
#pragma unroll
    for (int v = 0; v < 8; ++v) {
        int row0 = m0 + v + half * 8;
        int row1 = m0 + 16 + v + half * 8;
        O[(size_t)row0 * D + n0 + l15] = c0[v];
        O[(size_t)row1 * D + n0 + l15] = c1[v];
    }
}

// ---------------------------------------------------------------------------
extern "C" void kernel_launch(void* const* d_in, const int* in_sizes, int n_in,
                              void* d_out, int out_size, void* d_ws, size_t ws_size,
                              hipStream_t stream) {
    const float* seq   = (const float*)d_in[0];   // (n, L, d)
    const float* att   = (const float*)d_in[1];   // (n, h, L, L)
    const int*   pos   = (const int*)  d_in[2];   // (n, E, M)
    const float* mask  = (const float*)d_in[3];   // (n, E, M)
    const int*   hts   = (const int*)  d_in[4];   // (n, R, 2)
    float*       out   = (float*)d_out;           // (3, n*R, d)

    float* ws     = (float*)d_ws;
    float* e_emb  = ws;                                        // n*E*d      =    98,304 f
    float* e_att  = e_emb + (size_t)N_DOC * E * D;             // n*E*h*L    = 1,572,864 f
    float* ht_att = e_att + (size_t)N_DOC * E * H * L;         // n*R*L      = 1,048,576 f

    k_e_emb  <<<N_DOC * E,     256, 0, stream>>>(seq, pos, mask, e_emb);
    k_e_att  <<<N_DOC * E * H, 256, 0, stream>>>(att, pos, mask, e_att);
    k_ht_att <<<N_DOC * R,     256, 0, stream>>>(e_att, hts, ht_att);
    k_hs_ts  <<<N_DOC * R,     256, 0, stream>>>(e_emb, hts, out);
    k_rs_wmma<<<N_DOC * (R / 32) * (D / 16), 32, 0, stream>>>(seq, ht_att, out);
}